// RelativeDepthLoss_20074677141934
// MI455X (gfx1250) — compile-verified
//
#include <hip/hip_runtime.h>
#include <hip/hip_bf16.h>
#include <math.h>

// RelativeDepthLoss for MI455X (gfx1250, wave32).
// B=64, H=480, W=640, P=8192. Gather + masked softplus/square reduction.

typedef __attribute__((ext_vector_type(2))) float v2f;
typedef __attribute__((ext_vector_type(8))) float v8f;

#define BATCH 64
#define IMG_H 480
#define IMG_W 640
#define PAIRS 8192
#define SPLIT 8                         // blocks per batch image
#define TPB 256                         // 8 waves (wave32)
#define PAIRS_PER_BLOCK (PAIRS / SPLIT) // 1024
#define PPT (PAIRS_PER_BLOCK / TPB)     // 4 pairs per thread

// Kernel 1: per-block partial sums {sum_log, n_log, sum_sq, n_sq}.
// Hierarchy: per-thread regs -> wave32 xor-shuffle reduce -> LDS (8 waves x 4 vals)
// -> single v_wmma_f32_16x16x4_f32 (A = ones) collapses all 32 partials at once.
__global__ __launch_bounds__(TPB) void rdl_partial_kernel(
    const float* __restrict__ depth,
    const int* __restrict__ xA, const int* __restrict__ yA,
    const int* __restrict__ xB, const int* __restrict__ yB,
    const int* __restrict__ ord,
    float* __restrict__ ws)
{
  const int blk = blockIdx.x;
  const int b   = blk / SPLIT;
  const int s   = blk % SPLIT;
  const int t   = threadIdx.x;

  const float* __restrict__ img = depth + (size_t)b * (IMG_H * IMG_W);
  const size_t base = (size_t)b * PAIRS + (size_t)s * PAIRS_PER_BLOCK + (size_t)t * PPT;

  // Vectorized (b128) index loads: 16B-aligned since base is a multiple of 4.
  const int4 xa4 = *(const int4*)(xA  + base);
  const int4 ya4 = *(const int4*)(yA  + base);
  const int4 xb4 = *(const int4*)(xB  + base);
  const int4 yb4 = *(const int4*)(yB  + base);
  const int4 od4 = *(const int4*)(ord + base);

  const int xav[4] = {xa4.x, xa4.y, xa4.z, xa4.w};
  const int yav[4] = {ya4.x, ya4.y, ya4.z, ya4.w};
  const int xbv[4] = {xb4.x, xb4.y, xb4.z, xb4.w};
  const int ybv[4] = {yb4.x, yb4.y, yb4.z, yb4.w};
  const int odv[4] = {od4.x, od4.y, od4.z, od4.w};

  float sl = 0.f, nl = 0.f, sq = 0.f, ns = 0.f;
#pragma unroll
  for (int i = 0; i < PPT; ++i) {
    const float zA = img[xav[i] * IMG_W + yav[i]];
    const float zB = img[xbv[i] * IMG_W + ybv[i]];
    const float pred = zA - zB;
    const int o = odv[i];
    // numerically stable softplus(-t*pred) = max(x,0) + log1p(exp(-|x|)), x = -t*pred
    const float x  = -(float)o * pred;
    const float sp = fmaxf(x, 0.f) + log1pf(expf(-fabsf(x)));
    const bool valid = (o != 2);
    const bool islog = valid && (o != 0);
    const bool issq  = valid && (o == 0);
    sl += islog ? sp : 0.f;
    nl += islog ? 1.f : 0.f;
    sq += issq ? pred * pred : 0.f;
    ns += issq ? 1.f : 0.f;
  }

  // wave32 in-wave reduction: 5 xor-shuffle steps
#pragma unroll
  for (int m = 16; m >= 1; m >>= 1) {
    sl += __shfl_xor(sl, m, 32);
    nl += __shfl_xor(nl, m, 32);
    sq += __shfl_xor(sq, m, 32);
    ns += __shfl_xor(ns, m, 32);
  }

  __shared__ float part[4][8];  // [quantity][wave]
  const int lane = t & 31;
  const int wid  = t >> 5;
  if (lane == 0) {
    part[0][wid] = sl;
    part[1][wid] = nl;
    part[2][wid] = sq;
    part[3][wid] = ns;
  }
  __syncthreads();

  // Cross-wave stage via one WMMA. Pack the 32 partials into B (4x16 f32):
  //   B vgpr0: lanes 0-15 -> K=0, N=lane  (we use: quantity q = N/4, wave = N%4)
  //   B vgpr1: lanes 0-15 -> K=1, N=lane  (wave = N%4 + 4)
  //   lanes 16-31 (K=2,3) -> zero
  // With A = ones(16x4): D[m][n] = sum_k B[k][n] -> column sums. Columns 4q..4q+3
  // hold quantity q's pairwise wave sums; two xor-shuffles finish each quantity.
  // All 8 waves run the WMMA redundantly so EXEC is all-ones (ISA requirement).
  {
    const int L  = lane;
    const int q  = (L >> 2) & 3;
    const int w0 = L & 3;
    float b0 = (L < 16) ? part[q][w0]     : 0.f;
    float b1 = (L < 16) ? part[q][w0 + 4] : 0.f;

    v2f av;  av[0] = 1.f; av[1] = 1.f;
    v2f bvv; bvv[0] = b0; bvv[1] = b1;
    v8f c = {};
    c = __builtin_amdgcn_wmma_f32_16x16x4_f32(
        /*neg_a=*/false, av, /*neg_b=*/false, bvv,
        /*c_mod=*/(short)0, c, /*reuse_a=*/false, /*reuse_b=*/false);

    float col = c[0];                 // lane n (n<16): colsum_n; lanes 16-31 duplicate
    col += __shfl_xor(col, 1, 32);
    col += __shfl_xor(col, 2, 32);    // lanes 4q..4q+3 now hold total for quantity q

    const float t0 = __shfl(col, 0, 32);   // sum_log
    const float t1 = __shfl(col, 4, 32);   // n_log
    const float t2 = __shfl(col, 8, 32);   // sum_sq
    const float t3 = __shfl(col, 12, 32);  // n_sq

    if (t == 0) {
      float* w = ws + (size_t)blk * 4;
      w[0] = t0; w[1] = t1; w[2] = t2; w[3] = t3;
    }
  }
}

// Kernel 2: combine SPLIT block-partials per batch, normalize, mean over batch.
__global__ __launch_bounds__(64) void rdl_finalize_kernel(
    const float* __restrict__ ws, float* __restrict__ out)
{
  __shared__ float r[BATCH];
  const int t = threadIdx.x;  // 64 threads, one per batch image
  {
    float sl = 0.f, nl = 0.f, sq = 0.f, ns = 0.f;
#pragma unroll
    for (int s = 0; s < SPLIT; ++s) {
      const float* p = ws + (size_t)(t * SPLIT + s) * 4;
      sl += p[0]; nl += p[1]; sq += p[2]; ns += p[3];
    }
    r[t] = sl / nl + sq / ns;
  }
  __syncthreads();
  if (t < 32) {
    float v = r[t] + r[t + 32];
#pragma unroll
    for (int m = 16; m >= 1; m >>= 1) v += __shfl_xor(v, m, 32);
    if (t == 0) out[0] = v * (1.0f / (float)BATCH);
  }
}

extern "C" void kernel_launch(void* const* d_in, const int* in_sizes, int n_in,
                              void* d_out, int out_size, void* d_ws, size_t ws_size,
                              hipStream_t stream) {
  const float* depth = (const float*)d_in[0];
  const int*   xA    = (const int*)d_in[1];
  const int*   yA    = (const int*)d_in[2];
  const int*   xB    = (const int*)d_in[3];
  const int*   yB    = (const int*)d_in[4];
  const int*   ord   = (const int*)d_in[5];
  float* out = (float*)d_out;
  float* ws  = (float*)d_ws;   // needs BATCH*SPLIT*4 floats = 8 KB

  rdl_partial_kernel<<<BATCH * SPLIT, TPB, 0, stream>>>(depth, xA, yA, xB, yB, ord, ws);
  rdl_finalize_kernel<<<1, 64, 0, stream>>>(ws, out);
}